// _TorchWLKernel_14285061227092
// MI455X (gfx1250) — compile-verified
//
#include <hip/hip_runtime.h>
#include <stdint.h>

#define NB 64
#define NN 1024
#define NE 16384
#define NITER 5

typedef __attribute__((ext_vector_type(2))) float v2f;
typedef __attribute__((ext_vector_type(8))) float v8f;

// ---------------- zero fill ----------------
__global__ void k_zero(unsigned int* p, int n) {
  int i = blockIdx.x * blockDim.x + threadIdx.x;
  if (i < n) p[i] = 0u;
}

// ---------------- edge dedupe + degree (matches dense-mask semantics) ----------------
__global__ void k_dedupe(const int* __restrict__ src, const int* __restrict__ dst,
                         unsigned int* __restrict__ bits, int* __restrict__ deg,
                         unsigned char* __restrict__ valid) {
  int idx = blockIdx.x * blockDim.x + threadIdx.x;  // [0, NB*NE)
  if (idx >= NB * NE) return;
  int b = idx >> 14;                                 // NE = 2^14
  int s = src[idx], d = dst[idx];
  int key = (s << 10) | d;                           // N = 2^10
  unsigned int bit = 1u << (key & 31);
  unsigned int old = atomicOr(&bits[(b << 15) + (key >> 5)], bit);
  int ok = (old & bit) == 0u;
  valid[idx] = (unsigned char)ok;
  if (ok) atomicAdd(&deg[(b << 10) + s], 1);
}

// ---------------- copy initial labels + histogram them ----------------
__global__ void k_init(const int* __restrict__ labels_in, int* __restrict__ labels_cur,
                       float* __restrict__ feats) {
  int i = blockIdx.x * blockDim.x + threadIdx.x;     // [0, NB*NN)
  if (i >= NB * NN) return;
  int l = labels_in[i];
  labels_cur[i] = l;
  int b = i >> 10;
  atomicAdd(&feats[(b << 10) + l], 1.0f);
}

// ---------------- neighbor-label sum (integer atomics: exact, deterministic) ----------------
__global__ void k_seg(const int* __restrict__ src, const int* __restrict__ dst,
                      const unsigned char* __restrict__ valid,
                      const int* __restrict__ labels_cur, int* __restrict__ seg) {
  int idx = blockIdx.x * blockDim.x + threadIdx.x;
  if (idx >= NB * NE) return;
  if (!valid[idx]) return;
  int b = idx >> 14;
  atomicAdd(&seg[(b << 10) + src[idx]], labels_cur[(b << 10) + dst[idx]]);
}

// ---------------- WL hash + unique-inverse (dense rank) per graph ----------------
__global__ __launch_bounds__(1024) void k_wl(const int* __restrict__ deg,
                                             const int* __restrict__ seg,
                                             int* __restrict__ labels_cur,
                                             float* __restrict__ feats,
                                             const float* __restrict__ hw) {
  __shared__ float sh[NN];
  __shared__ int fo[NN];
  int b = blockIdx.x, i = threadIdx.x;
  float w0 = hw[0], w1 = hw[1];
  float degf = (float)deg[(b << 10) + i];
  sh[i] = degf;
  __syncthreads();
  for (int s = 512; s > 0; s >>= 1) {
    if (i < s) sh[i] = fmaxf(sh[i], sh[i + s]);
    __syncthreads();
  }
  float K = sh[0];
  __syncthreads();
  float lab  = (float)labels_cur[(b << 10) + i];
  float segf = (float)seg[(b << 10) + i];
  // Match JAX op order exactly: (K*w0)*lab + w1*((seg+deg)-K), no FMA contraction.
  float h = __fadd_rn(__fmul_rn(__fmul_rn(K, w0), lab),
                      __fmul_rn(w1, __fsub_rn(__fadd_rn(segf, degf), K)));
  sh[i] = h;
  __syncthreads();
  // first occurrence flag (any representative of a tie group works)
  int first = 1;
  for (int j = 0; j < i; ++j) {
    if (sh[j] == h) { first = 0; break; }
  }
  fo[i] = first;
  __syncthreads();
  // dense rank = number of distinct values strictly less than h
  int rank = 0;
  for (int j = 0; j < NN; ++j) rank += (fo[j] && (sh[j] < h)) ? 1 : 0;
  labels_cur[(b << 10) + i] = rank;
  atomicAdd(&feats[(b << 10) + rank], 1.0f);
}

// ---------------- Gram matrix via V_WMMA_F32_16X16X4_F32 + normalization ----------------
__global__ __launch_bounds__(512) void k_gram(const float* __restrict__ feats,
                                              float* __restrict__ out) {
  __shared__ float gram[64 * 64];
  __shared__ float dsc[64];
  int tid = threadIdx.x;
  int wave = tid >> 5, lane = tid & 31;
  int tr = wave >> 2, tc = wave & 3;      // 4x4 grid of 16x16 output tiles
  int l16 = lane & 15;
  int hi = lane >> 4;                     // 0: K=0,1  1: K=2,3 (ISA 16x4 f32 A layout)
  const float* arow = feats + (((tr * 16 + l16) << 10) + hi * 2);
  const float* brow = feats + (((tc * 16 + l16) << 10) + hi * 2);
  v8f acc = {0.f, 0.f, 0.f, 0.f, 0.f, 0.f, 0.f, 0.f};
  for (int k = 0; k < NN; k += 4) {
    v2f a = *(const v2f*)(arow + k);
    v2f bb = *(const v2f*)(brow + k);
    acc = __builtin_amdgcn_wmma_f32_16x16x4_f32(false, a, false, bb,
                                                (short)0, acc, false, false);
  }
  // C/D layout: VGPR r -> row r (lanes 0-15) / row r+8 (lanes 16-31), col = lane%16
  int col = tc * 16 + l16;
  int rbase = tr * 16 + hi * 8;
  for (int r = 0; r < 8; ++r) gram[(rbase + r) * 64 + col] = acc[r];
  __syncthreads();
  if (tid < 64) dsc[tid] = sqrtf(gram[tid * 64 + tid]);
  __syncthreads();
  for (int idx = tid; idx < 64 * 64; idx += 512) {
    int ii = idx >> 6, jj = idx & 63;
    out[idx] = gram[idx] / (dsc[jj] * dsc[ii]);
  }
}

extern "C" void kernel_launch(void* const* d_in, const int* in_sizes, int n_in,
                              void* d_out, int out_size, void* d_ws, size_t ws_size,
                              hipStream_t stream) {
  const int* e_src      = (const int*)d_in[0];
  const int* e_dst      = (const int*)d_in[1];
  const int* labels_in  = (const int*)d_in[2];
  const float* hw       = (const float*)d_in[3];

  char* ws = (char*)d_ws;
  // layout: bits(8MB) | deg(256KB) | feats(256KB) | seg(256KB) | labels_cur(256KB) | valid(1MB)
  unsigned int* bits   = (unsigned int*)(ws);
  int*          deg    = (int*)(ws + 8388608);
  float*        feats  = (float*)(ws + 8650752);
  int*          seg    = (int*)(ws + 8912896);
  int*          labels = (int*)(ws + 9175040);
  unsigned char* valid = (unsigned char*)(ws + 9437184);

  // zero bits + deg + feats (contiguous region)
  int zwords = (8388608 + 262144 + 262144) / 4;
  k_zero<<<(zwords + 255) / 256, 256, 0, stream>>>(bits, zwords);
  k_dedupe<<<(NB * NE + 255) / 256, 256, 0, stream>>>(e_src, e_dst, bits, deg, valid);
  k_init<<<(NB * NN + 255) / 256, 256, 0, stream>>>(labels_in, labels, feats);

  for (int it = 0; it < NITER; ++it) {
    k_zero<<<(NB * NN + 255) / 256, 256, 0, stream>>>((unsigned int*)seg, NB * NN);
    k_seg<<<(NB * NE + 255) / 256, 256, 0, stream>>>(e_src, e_dst, valid, labels, seg);
    k_wl<<<NB, NN, 0, stream>>>(deg, seg, labels, feats, hw);
  }

  k_gram<<<1, 512, 0, stream>>>(feats, (float*)d_out);
}